// GNNModel_48490180771963
// MI455X (gfx1250) — compile-verified
//
#include <hip/hip_runtime.h>
#include <hip/hip_bf16.h>
#include <stdint.h>

typedef __attribute__((ext_vector_type(2))) float v2f;
typedef __attribute__((ext_vector_type(8))) float v8f;

#define DH 128

// ---------------- degree / normalization ----------------
__global__ void fill_one(float* __restrict__ p, int n) {
    int i = blockIdx.x * blockDim.x + threadIdx.x;
    if (i < n) p[i] = 1.0f;   // self-loop contributes 1 to every node's degree
}

__global__ void deg_count(const int* __restrict__ dst, float* __restrict__ deg, int E) {
    int e = blockIdx.x * blockDim.x + threadIdx.x;
    if (e < E) atomicAdd(&deg[dst[e]], 1.0f);
}

__global__ void deg_rsqrt(float* __restrict__ p, int n) {
    int i = blockIdx.x * blockDim.x + threadIdx.x;
    if (i < n) p[i] = rsqrtf(p[i]);
}

// ---------------- WMMA f32 GEMM: T = H(NxK=128) @ W(128x128) ----------------
// Block: 16 rows x 128 cols, 256 threads = 8 waves; wave w computes col tile w*16.
// A tile staged in LDS (pad stride 132 -> conflict free: bank = (4m + c) % 64).
__global__ void __launch_bounds__(256) gemm_wmma(const float* __restrict__ H,
                                                 const float* __restrict__ W,
                                                 float* __restrict__ T, int N) {
    __shared__ float As[16 * 132];
    int row_base = blockIdx.x * 16;
    for (int i4 = threadIdx.x; i4 < 512; i4 += 256) {   // 16x128 floats as float4
        int r  = i4 >> 5;
        int c4 = (i4 & 31) << 2;
        int rr = row_base + r; if (rr >= N) rr = N - 1;
        float4 v = *(const float4*)(H + (size_t)rr * DH + c4);
        *(float4*)(&As[r * 132 + c4]) = v;              // 132*4B = 528B, 16B aligned
    }
    __syncthreads();

    int wave = threadIdx.x >> 5;
    int lane = threadIdx.x & 31;
    int m    = lane & 15;
    int hi   = lane >> 4;              // 0: lanes 0-15 (K+0,K+1), 1: lanes 16-31 (K+2,K+3)
    int col  = (wave << 4) + m;

    v8f acc = {};
    const float* arow = &As[m * 132];
    #pragma unroll
    for (int k0 = 0; k0 < DH; k0 += 4) {
        int ka = k0 + 2 * hi;
        v2f a, b;
        a.x = arow[ka];
        a.y = arow[ka + 1];
        b.x = W[(size_t)ka       * DH + col];
        b.y = W[(size_t)(ka + 1) * DH + col];
        acc = __builtin_amdgcn_wmma_f32_16x16x4_f32(false, a, false, b,
                                                    (short)0, acc, false, false);
    }
    #pragma unroll
    for (int j = 0; j < 8; ++j) {      // C/D layout: VGPR j -> rows j (lo half), j+8 (hi half)
        int rr = row_base + j + 8 * hi;
        if (rr < N) T[(size_t)rr * DH + col] = acc[j];
    }
}

// ---------------- aggregation ----------------
// O[i,:] = T[i,:] * dinv[i]^2   (self-loop term, also zero-initializes O)
__global__ void self_init(const float* __restrict__ T, const float* __restrict__ dinv,
                          float* __restrict__ O, int N) {
    int t = blockIdx.x * blockDim.x + threadIdx.x;   // over N*32 float4s
    if (t >= N * 32) return;
    int node = t >> 5;
    float s = dinv[node]; s = s * s;
    float4 v = ((const float4*)T)[t];
    v.x *= s; v.y *= s; v.z *= s; v.w *= s;
    ((float4*)O)[t] = v;
}

// one wave per edge: O[dst,:] += T[src,:] * dinv[src]*dinv[dst]
__global__ void __launch_bounds__(256) edge_scatter(const float* __restrict__ T,
                                                    const int* __restrict__ src,
                                                    const int* __restrict__ dst,
                                                    const float* __restrict__ dinv,
                                                    float* __restrict__ O, int E) {
    int e = blockIdx.x * 8 + (threadIdx.x >> 5);
    if (e >= E) return;
    int lane = threadIdx.x & 31;
    int s = src[e];
    int d = dst[e];
    float nrm = dinv[s] * dinv[d];
    float4 v = *(const float4*)(T + (size_t)s * DH + (lane << 2));
    float* o = O + (size_t)d * DH + (lane << 2);
    atomicAdd(o + 0, v.x * nrm);
    atomicAdd(o + 1, v.y * nrm);
    atomicAdd(o + 2, v.z * nrm);
    atomicAdd(o + 3, v.w * nrm);
}

__global__ void bias_act(float* __restrict__ O, const float* __restrict__ b,
                         int N, int relu) {
    int t = blockIdx.x * blockDim.x + threadIdx.x;   // over N*32 float4s
    if (t >= N * 32) return;
    int q = t & 31;
    float4 bb = ((const float4*)b)[q];
    float4 v  = ((float4*)O)[t];
    v.x += bb.x; v.y += bb.y; v.z += bb.z; v.w += bb.w;
    if (relu) {
        v.x = fmaxf(v.x, 0.0f); v.y = fmaxf(v.y, 0.0f);
        v.z = fmaxf(v.z, 0.0f); v.w = fmaxf(v.w, 0.0f);
    }
    ((float4*)O)[t] = v;
}

// ---------------- final matvec: out[i] = H[i,:] . w + b ----------------
__global__ void __launch_bounds__(256) matvec_out(const float* __restrict__ H,
                                                  const float* __restrict__ w,
                                                  const float* __restrict__ b,
                                                  float* __restrict__ out, int N) {
    int node = blockIdx.x * 8 + (threadIdx.x >> 5);
    if (node >= N) return;
    int lane = threadIdx.x & 31;
    float4 h  = *(const float4*)(H + (size_t)node * DH + (lane << 2));
    float4 ww = *(const float4*)(w + (lane << 2));
    float s = h.x * ww.x + h.y * ww.y + h.z * ww.z + h.w * ww.w;
    #pragma unroll
    for (int off = 16; off > 0; off >>= 1) s += __shfl_xor(s, off, 32);
    if (lane == 0) out[node] = s + b[0];
}

extern "C" void kernel_launch(void* const* d_in, const int* in_sizes, int n_in,
                              void* d_out, int out_size, void* d_ws, size_t ws_size,
                              hipStream_t stream) {
    const float* x      = (const float*)d_in[0];
    const int*   ei     = (const int*)  d_in[1];   // int32 (JAX default x64-off)
    const float* W0     = (const float*)d_in[2];
    const float* b0     = (const float*)d_in[3];
    const float* W1     = (const float*)d_in[4];
    const float* b1     = (const float*)d_in[5];
    const float* W2     = (const float*)d_in[6];
    const float* b2     = (const float*)d_in[7];
    const float* mw1    = (const float*)d_in[8];
    const float* mb1    = (const float*)d_in[9];
    const float* mw2    = (const float*)d_in[10];
    const float* mb2    = (const float*)d_in[11];

    int N = in_sizes[0] / DH;     // 50000
    int E = in_sizes[1] / 2;      // 600000
    const int* srcI = ei;         // edge_index[0]
    const int* dstI = ei + E;     // edge_index[1]

    float* dinv = (float*)d_ws;
    size_t off  = (((size_t)N * 4) + 255) & ~(size_t)255;
    float* bufA = (float*)((char*)d_ws + off);
    float* bufB = bufA + (size_t)N * DH;

    int tN   = (N + 255) / 256;
    int tE   = (E + 255) / 256;
    int tN32 = (N * 32 + 255) / 256;
    int gemmG = (N + 15) / 16;
    int edgeG = (E + 7) / 8;
    int nodeG = (N + 7) / 8;

    // normalization coefficients
    fill_one <<<tN, 256, 0, stream>>>(dinv, N);
    deg_count<<<tE, 256, 0, stream>>>(dstI, dinv, E);
    deg_rsqrt<<<tN, 256, 0, stream>>>(dinv, N);

    // layer 1: x -> bufB (gemm) -> bufA (agg)
    gemm_wmma   <<<gemmG, 256, 0, stream>>>(x, W0, bufB, N);
    self_init   <<<tN32,  256, 0, stream>>>(bufB, dinv, bufA, N);
    edge_scatter<<<edgeG, 256, 0, stream>>>(bufB, srcI, dstI, dinv, bufA, E);
    bias_act    <<<tN32,  256, 0, stream>>>(bufA, b0, N, 1);

    // layer 2
    gemm_wmma   <<<gemmG, 256, 0, stream>>>(bufA, W1, bufB, N);
    self_init   <<<tN32,  256, 0, stream>>>(bufB, dinv, bufA, N);
    edge_scatter<<<edgeG, 256, 0, stream>>>(bufB, srcI, dstI, dinv, bufA, E);
    bias_act    <<<tN32,  256, 0, stream>>>(bufA, b1, N, 1);

    // layer 3 (no ReLU)
    gemm_wmma   <<<gemmG, 256, 0, stream>>>(bufA, W2, bufB, N);
    self_init   <<<tN32,  256, 0, stream>>>(bufB, dinv, bufA, N);
    edge_scatter<<<edgeG, 256, 0, stream>>>(bufB, srcI, dstI, dinv, bufA, E);
    bias_act    <<<tN32,  256, 0, stream>>>(bufA, b2, N, 0);

    // MLP head
    gemm_wmma   <<<gemmG, 256, 0, stream>>>(bufA, mw1, bufB, N);
    bias_act    <<<tN32,  256, 0, stream>>>(bufB, mb1, N, 1);
    matvec_out  <<<nodeG, 256, 0, stream>>>(bufB, mw2, mb2, (float*)d_out, N);
}